// CombinedGeneModel_72610717106295
// MI455X (gfx1250) — compile-verified
//
#include <hip/hip_runtime.h>
#include <cstdint>
#include <cstddef>

// Problem constants (match reference).
#define N_GENES 20000
#define N_TECH  2
#define N_EXP   4
#define BATCH_N 1024

typedef float f4 __attribute__((ext_vector_type(4)));

constexpr int kThreads       = 128;
constexpr int kGenesPerThr   = 4;
constexpr int kGenesPerBlock = kThreads * kGenesPerThr;                       // 512
constexpr int kGeneBlocks    = (N_GENES + kGenesPerBlock - 1) / kGenesPerBlock; // 40
constexpr int kBatchTiles    = 32;
constexpr int kBatchPerTile  = BATCH_N / kBatchTiles;                         // 32
constexpr int kSmallChunks   = 5;  // b2(t0), b2(t1), wg(g..g+1), wg(g+2..g+3), bg

// CDNA5 async copy: 16 bytes global -> LDS, tracked by ASYNCcnt (ISA 15.18.3 op 98).
// GV addressing form: vdst = per-lane LDS byte offset, vaddr = 64-bit global address.
__device__ __forceinline__ void async_copy_b128(uint32_t lds_off, const void* gaddr) {
  asm volatile("global_load_async_to_lds_b128 %0, %1, off"
               :: "v"(lds_off), "v"(gaddr)
               : "memory");
}

__device__ __forceinline__ float bellows1(float xv, f4 w1v, f4 b1v, f4 w2v, float b2v) {
  float acc = b2v;
#pragma unroll
  for (int e = 0; e < N_EXP; ++e) {
    float h = fmaxf(fmaf(xv, w1v[e], b1v[e]), 0.0f);   // expand + relu
    acc = fmaf(h, w2v[e], acc);                        // shrink
  }
  return fmaxf(acc, 0.0f);                             // relu
}

__global__ __launch_bounds__(kThreads)
void gene_mlp_kernel(const float* __restrict__ x,    // [B, T, G]
                     const float* __restrict__ w1,   // [R=T*G, E]
                     const float* __restrict__ b1,   // [R, E]
                     const float* __restrict__ w2,   // [R, E]
                     const float* __restrict__ b2,   // [R]
                     const float* __restrict__ wg,   // [G, T]
                     const float* __restrict__ bg,   // [G]
                     float* __restrict__ out) {      // [B, G]
  __shared__ f4 stage[kThreads * kSmallChunks];      // 10 KB, never occupancy-limiting

  const int g = (blockIdx.x * kThreads + threadIdx.x) * kGenesPerThr;
  if (g >= N_GENES) return;   // G % 4 == 0, so a thread is all-valid or all-invalid

  // ---- One-time async-tensor staging of the small per-gene params (ASYNCcnt path).
  // Low 32 bits of a flat LDS address are the WG-relative LDS byte offset (ISA 10.2).
  const uint32_t lbase = (uint32_t)(uintptr_t)&stage[threadIdx.x * kSmallChunks];
  async_copy_b128(lbase + 0 * 16, b2 + g);                          // b2, tech0, genes g..g+3
  async_copy_b128(lbase + 1 * 16, b2 + N_GENES + g);                // b2, tech1
  async_copy_b128(lbase + 2 * 16, wg + (size_t)g * N_TECH);         // wg for genes g, g+1
  async_copy_b128(lbase + 3 * 16, wg + (size_t)(g + 2) * N_TECH);   // wg for genes g+2, g+3
  async_copy_b128(lbase + 4 * 16, bg + g);                          // bias_g

  // ---- Big per-gene weights straight into registers (b128 loads, L2-resident,
  // reused across all 1024 batches -> weight HBM traffic is negligible).
  f4 w1t0[4], b1t0[4], w2t0[4], w1t1[4], b1t1[4], w2t1[4];
#pragma unroll
  for (int i = 0; i < 4; ++i) {
    const size_t r0 = (size_t)(g + i) * N_EXP;            // row for tech0: r = g+i
    const size_t r1 = (size_t)(N_GENES + g + i) * N_EXP;  // row for tech1: r = G+g+i
    w1t0[i] = *(const f4*)(w1 + r0);
    b1t0[i] = *(const f4*)(b1 + r0);
    w2t0[i] = *(const f4*)(w2 + r0);
    w1t1[i] = *(const f4*)(w1 + r1);
    b1t1[i] = *(const f4*)(b1 + r1);
    w2t1[i] = *(const f4*)(w2 + r1);
  }

  // Drain the async copies for this wave, then pull staged params into registers.
  asm volatile("s_wait_asynccnt 0" ::: "memory");
  const f4* tw = &stage[threadIdx.x * kSmallChunks];
  const f4 b2t0 = tw[0];
  const f4 b2t1 = tw[1];
  const f4 wgA  = tw[2];   // {wg[g][0], wg[g][1], wg[g+1][0], wg[g+1][1]}
  const f4 wgB  = tw[3];   // same for genes g+2, g+3
  const f4 bgv  = tw[4];

  // ---- Streaming loop over this tile's batches: 2 b128 loads + 1 b128 store each.
  const int b0 = blockIdx.y * kBatchPerTile;
  constexpr size_t xstride = (size_t)N_TECH * N_GENES;
  const float* xp = x   + (size_t)b0 * xstride + g;
  float*       op = out + (size_t)b0 * N_GENES + g;

  for (int bi = 0; bi < kBatchPerTile; ++bi) {
    if (bi + 1 < kBatchPerTile) {
      // CDNA5 global_prefetch_b8: stream-ahead next batch's x rows.
      __builtin_prefetch(xp + xstride, 0, 0);
      __builtin_prefetch(xp + xstride + N_GENES, 0, 0);
    }
    const f4 x0 = *(const f4*)(xp);              // x[b, tech0, g..g+3]
    const f4 x1 = *(const f4*)(xp + N_GENES);    // x[b, tech1, g..g+3]
    f4 res{};
#pragma unroll
    for (int i = 0; i < 4; ++i) {
      const float s0 = bellows1(x0[i], w1t0[i], b1t0[i], w2t0[i], b2t0[i]);
      const float s1 = bellows1(x1[i], w1t1[i], b1t1[i], w2t1[i], b2t1[i]);
      const f4 wgv = (i < 2) ? wgA : wgB;
      const float wt0 = wgv[(i & 1) * 2 + 0];
      const float wt1 = wgv[(i & 1) * 2 + 1];
      res[i] = fmaxf(fmaf(s1, wt1, fmaf(s0, wt0, bgv[i])), 0.0f);
    }
    *(f4*)op = res;
    xp += xstride;
    op += N_GENES;
  }
}

extern "C" void kernel_launch(void* const* d_in, const int* in_sizes, int n_in,
                              void* d_out, int out_size, void* d_ws, size_t ws_size,
                              hipStream_t stream) {
  (void)in_sizes; (void)n_in; (void)out_size; (void)d_ws; (void)ws_size;
  const float* x  = (const float*)d_in[0];   // [1024, 2, 20000]
  const float* w1 = (const float*)d_in[1];   // [40000, 4]
  const float* b1 = (const float*)d_in[2];   // [40000, 4]
  const float* w2 = (const float*)d_in[3];   // [40000, 4]
  const float* b2 = (const float*)d_in[4];   // [40000]
  const float* wg = (const float*)d_in[5];   // [20000, 2]
  const float* bg = (const float*)d_in[6];   // [20000]
  float* out = (float*)d_out;                // [1024, 20000]

  dim3 grid(kGeneBlocks, kBatchTiles);       // 40 x 32 = 1280 blocks
  gene_mlp_kernel<<<grid, kThreads, 0, stream>>>(x, w1, b1, w2, b2, wg, bg, out);
}